// global_local_fusion_17231408791842
// MI455X (gfx1250) — compile-verified
//
#include <hip/hip_runtime.h>
#include <stddef.h>

// ---------------------------------------------------------------------------
// CDNA5 / gfx1250 fused global-local-fusion block, all GEMMs via
// V_WMMA_F32_16X16X4_F32 (native f32 matrix path, wave32).
//
// Layout convention (per CDNA5 ISA, 32-bit matrices):
//   A (16x4):  lane l holds A[M=l%16][K=koff+j] in VGPR j,  koff = 2*(l>>4)
//   B (4x16):  lane l holds B[K=koff+j][N=l%16] in VGPR j   (mirrored layout)
//   C/D:       VGPR r, lane l -> D[M=r+8*(l>>4)][N=l%16]
// ---------------------------------------------------------------------------

typedef float v2f __attribute__((ext_vector_type(2)));
typedef float v8f __attribute__((ext_vector_type(8)));

#define WMMA_F32X4(A, Bv, Cv) \
  __builtin_amdgcn_wmma_f32_16x16x4_f32(false, (A), false, (Bv), (short)0, (Cv), false, false)

static constexpr int IMG_W  = 256;
static constexpr int IMG_HW = 65536;   // 256*256

// ---------------------------------------------------------------------------
// K1: cat = conv1x1_{128->64}( concat(pixel_shuffle(high,2), low) )
// one wave = 16 contiguous pixels x 64 outputs ; K = 128 in 32 wmma steps x 4 ntiles
// ---------------------------------------------------------------------------
__global__ __launch_bounds__(256) void k1_reduce(
    const float* __restrict__ high, const float* __restrict__ low,
    const float* __restrict__ rw,   const float* __restrict__ rb,
    float* __restrict__ cat)
{
  __shared__ float lw[64 * 132];                       // reduce_w [o][c], padded
  for (int i = threadIdx.x; i < 64 * 128; i += 256)
    lw[(i >> 7) * 132 + (i & 127)] = rw[i];
  __syncthreads();

  const int lane = threadIdx.x & 31;
  const int tile = blockIdx.x * 8 + (threadIdx.x >> 5);
  const int b    = tile >> 12;                         // 4096 tiles per image
  const int p0   = (tile & 4095) << 4;
  const int h    = p0 >> 8;
  const int w0   = p0 & 255;
  const int n    = lane & 15;
  const int koff = (lane >> 4) << 1;

  const int wc = w0 + n;                               // this lane's pixel column (M = n)
  // pixel_shuffle(2): ps[b,c,H,W] = high[b, 4c + 2*(H&1) + (W&1), H/2, W/2]
  const float* aHigh = high
      + (((size_t)b * 256 + (size_t)((h & 1) * 2 + (wc & 1))) * 128 + (h >> 1)) * 128
      + (wc >> 1);                                     // stride per cat-channel: 4*128*128
  const float* aLow  = low + (size_t)b * 64 * IMG_HW + (size_t)h * IMG_W + wc;

  v8f acc0 = {}, acc1 = {}, acc2 = {}, acc3 = {};
  for (int kk = 0; kk < 128; kk += 4) {
    const int c0 = kk + koff;
    v2f a;
    if (kk < 64) {                                     // uniform branch per k-step
      a.x = aHigh[(size_t)c0 * 65536];
      a.y = aHigh[(size_t)(c0 + 1) * 65536];
    } else {
      a.x = aLow[(size_t)(c0 - 64) * IMG_HW];
      a.y = aLow[(size_t)(c0 - 63) * IMG_HW];
    }
    v2f b0, b1, b2, b3;
    b0.x = lw[n * 132 + c0];         b0.y = lw[n * 132 + c0 + 1];
    b1.x = lw[(n + 16) * 132 + c0];  b1.y = lw[(n + 16) * 132 + c0 + 1];
    b2.x = lw[(n + 32) * 132 + c0];  b2.y = lw[(n + 32) * 132 + c0 + 1];
    b3.x = lw[(n + 48) * 132 + c0];  b3.y = lw[(n + 48) * 132 + c0 + 1];
    acc0 = WMMA_F32X4(a, b0, acc0);
    acc1 = WMMA_F32X4(a, b1, acc1);
    acc2 = WMMA_F32X4(a, b2, acc2);
    acc3 = WMMA_F32X4(a, b3, acc3);
  }

  const int mb = (lane >> 4) << 3;                     // M base of this lane's D rows
  float* outp = cat + (size_t)b * 64 * IMG_HW + p0;
#define K1_STORE(NT, ACC)                                                    \
  {                                                                          \
    const int o = (NT) * 16 + n;                                             \
    const float bias = rb[o];                                                \
    _Pragma("unroll") for (int r = 0; r < 8; ++r)                            \
        outp[(size_t)o * IMG_HW + mb + r] = (ACC)[r] + bias;                 \
  }
  K1_STORE(0, acc0) K1_STORE(1, acc1) K1_STORE(2, acc2) K1_STORE(3, acc3)
#undef K1_STORE
}

// ---------------------------------------------------------------------------
// K2: pre = lrelu( conv1x1_{64->16}(cat) + conv1x1_{256->16}(emb) )
// combined GEMM with K = 64+256 = 320, N = 16
// ---------------------------------------------------------------------------
__global__ __launch_bounds__(256) void k2_pre(
    const float* __restrict__ cat, const float* __restrict__ emb,
    const float* __restrict__ w1,  const float* __restrict__ b1v,
    const float* __restrict__ we,  const float* __restrict__ bev,
    float* __restrict__ pre)
{
  __shared__ float lw[16 * 321];                       // [o][k] combined weights
  for (int i = threadIdx.x; i < 16 * 320; i += 256) {
    const int o = i / 320, c = i % 320;
    lw[o * 321 + c] = (c < 64) ? w1[o * 64 + c] : we[o * 256 + (c - 64)];
  }
  __syncthreads();

  const int lane = threadIdx.x & 31;
  const int tile = blockIdx.x * 8 + (threadIdx.x >> 5);
  const int b    = tile >> 12;
  const int p0   = (tile & 4095) << 4;
  const int n    = lane & 15;
  const int koff = (lane >> 4) << 1;

  const float* aCat = cat + (size_t)b * 64  * IMG_HW + p0 + n;
  const float* aEmb = emb + (size_t)b * 256 * IMG_HW + p0 + n;

  v8f acc = {};
  for (int kk = 0; kk < 320; kk += 4) {
    const int c0 = kk + koff;
    v2f a;
    if (kk < 64) {
      a.x = aCat[(size_t)c0 * IMG_HW];
      a.y = aCat[(size_t)(c0 + 1) * IMG_HW];
    } else {
      a.x = aEmb[(size_t)(c0 - 64) * IMG_HW];
      a.y = aEmb[(size_t)(c0 - 63) * IMG_HW];
    }
    v2f bv;
    bv.x = lw[n * 321 + c0];
    bv.y = lw[n * 321 + c0 + 1];
    acc = WMMA_F32X4(a, bv, acc);
  }

  const int mb = (lane >> 4) << 3;
  const float bias = b1v[n] + bev[n];
  float* outp = pre + ((size_t)b * 16 + n) * IMG_HW + p0 + mb;
  #pragma unroll
  for (int r = 0; r < 8; ++r) {
    float v = acc[r] + bias;
    outp[r] = (v > 0.f) ? v : 0.01f * v;               // LeakyReLU fused here
  }
}

// ---------------------------------------------------------------------------
// K3: weight = conv3x3_{16->36}(pre)   (im2col GEMM, K = 16*9 = 144, N padded 48)
// ---------------------------------------------------------------------------
__global__ __launch_bounds__(256) void k3_conv3x3(
    const float* __restrict__ pre, const float* __restrict__ w2,
    const float* __restrict__ b2v, float* __restrict__ wgt)
{
  __shared__ float lw[144 * 48];                       // [k][o], o padded to 48
  for (int i = threadIdx.x; i < 144 * 48; i += 256) {
    const int k = i / 48, o = i % 48;
    const int ci = k / 9, r9 = k % 9, ky = r9 / 3, kx = r9 % 3;
    lw[i] = (o < 36) ? w2[((o * 16 + ci) * 3 + ky) * 3 + kx] : 0.0f;
  }
  __syncthreads();

  const int lane = threadIdx.x & 31;
  const int tile = blockIdx.x * 8 + (threadIdx.x >> 5);
  const int b    = tile >> 12;
  const int p0   = (tile & 4095) << 4;
  const int h    = p0 >> 8;
  const int w0   = p0 & 255;
  const int n    = lane & 15;
  const int koff = (lane >> 4) << 1;
  const int wc   = w0 + n;

  const float* preB = pre + (size_t)b * 16 * IMG_HW;
  v8f acc0 = {}, acc1 = {}, acc2 = {};
  for (int kk = 0; kk < 144; kk += 4) {
    const int c0 = kk + koff;
    v2f a;
    {
      const int ci = c0 / 9, r9 = c0 % 9;
      const int hh = h + r9 / 3 - 1, ww = wc + r9 % 3 - 1;
      a.x = (hh >= 0 && hh < 256 && ww >= 0 && ww < 256)
              ? preB[(size_t)ci * IMG_HW + hh * IMG_W + ww] : 0.f;
    }
    {
      const int c1 = c0 + 1;
      const int ci = c1 / 9, r9 = c1 % 9;
      const int hh = h + r9 / 3 - 1, ww = wc + r9 % 3 - 1;
      a.y = (hh >= 0 && hh < 256 && ww >= 0 && ww < 256)
              ? preB[(size_t)ci * IMG_HW + hh * IMG_W + ww] : 0.f;
    }
    v2f b0, b1, b2;
    b0.x = lw[c0 * 48 + n];       b0.y = lw[(c0 + 1) * 48 + n];
    b1.x = lw[c0 * 48 + 16 + n];  b1.y = lw[(c0 + 1) * 48 + 16 + n];
    b2.x = lw[c0 * 48 + 32 + n];  b2.y = lw[(c0 + 1) * 48 + 32 + n];
    acc0 = WMMA_F32X4(a, b0, acc0);
    acc1 = WMMA_F32X4(a, b1, acc1);
    acc2 = WMMA_F32X4(a, b2, acc2);
  }

  const int mb = (lane >> 4) << 3;
  float* outp = wgt + (size_t)b * 36 * IMG_HW + p0 + mb;
#define K3_STORE(NT, ACC)                                                    \
  {                                                                          \
    const int o = (NT) * 16 + n;                                             \
    if (o < 36) {                                                            \
      const float bias = b2v[o];                                             \
      _Pragma("unroll") for (int r = 0; r < 8; ++r)                          \
          outp[(size_t)o * IMG_HW + r] = (ACC)[r] + bias;                    \
    }                                                                        \
  }
  K3_STORE(0, acc0) K3_STORE(1, acc1) K3_STORE(2, acc2)
#undef K3_STORE
}

// ---------------------------------------------------------------------------
// K4: dynamic 3x3 group filtering (per-pixel weights, VALU) producing the
// out-conv A operand *directly in WMMA A-register layout*, then
// conv1x1_{64->32} + BN(folded) + LeakyReLU via 2x16 WMMA steps.
// Lane l handles pixel M=l%16 and the 32 channels with c%4 in {2*half, 2*half+1}.
// ---------------------------------------------------------------------------
__global__ __launch_bounds__(256) void k4_final(
    const float* __restrict__ cat, const float* __restrict__ wgt,
    const float* __restrict__ ow,  const float* __restrict__ ob,
    const float* __restrict__ gmm, const float* __restrict__ bta,
    const float* __restrict__ mu,  const float* __restrict__ var,
    float* __restrict__ out)
{
  __shared__ float lw[32 * 65];                        // out_w * bn_scale, [o][c]
  __shared__ float lsh[32];                            // folded bias/shift
  for (int i = threadIdx.x; i < 32 * 64; i += 256) {
    const int o = i >> 6;
    const float sc = gmm[o] * rsqrtf(var[o] + 1e-5f);
    lw[o * 65 + (i & 63)] = ow[i] * sc;
  }
  if (threadIdx.x < 32) {
    const int o = threadIdx.x;
    const float sc = gmm[o] * rsqrtf(var[o] + 1e-5f);
    lsh[o] = (ob[o] - mu[o]) * sc + bta[o];
  }
  __syncthreads();

  const int lane = threadIdx.x & 31;
  const int tile = blockIdx.x * 8 + (threadIdx.x >> 5);
  const int b    = tile >> 12;
  const int p0   = (tile & 4095) << 4;
  const int h    = p0 >> 8;
  const int w0   = p0 & 255;
  const int n    = lane & 15;
  const int half = lane >> 4;
  const int wc   = w0 + n;

  // 36 per-pixel dynamic filter taps for this lane's pixel
  float wk[36];
  const float* wgtp = wgt + (size_t)b * 36 * IMG_HW + p0 + n;
  #pragma unroll
  for (int i = 0; i < 36; ++i) wk[i] = wgtp[(size_t)i * IMG_HW];

  float av[32];
  #pragma unroll
  for (int i = 0; i < 32; ++i) av[i] = 0.f;

  const float* catB = cat + (size_t)b * 64 * IMG_HW;
  #pragma unroll
  for (int k = 0; k < 9; ++k) {                        // tap k = (dy+1)*3 + (dx+1)
    const int hh = h + k / 3 - 1, ww = wc + k % 3 - 1;
    const bool ok = (hh >= 0) && (hh < 256) && (ww >= 0) && (ww < 256);
    const float* src = catB + (ptrdiff_t)hh * IMG_W + ww;
    #pragma unroll
    for (int t = 0; t < 16; ++t) {
      const int c = 4 * t + 2 * half;                  // channel pair (group g = c>>4)
      const float dw = wk[(c >> 4) * 9 + k];
      const float x0 = ok ? src[(size_t)c * IMG_HW] : 0.f;
      const float x1 = ok ? src[(size_t)(c + 1) * IMG_HW] : 0.f;
      av[2 * t]     += dw * x0;
      av[2 * t + 1] += dw * x1;
    }
  }

  // out conv 64->32: av[] already sits in A layout (K = 4t + 2*half + j)
  v8f acc0 = {}, acc1 = {};
  #pragma unroll
  for (int t = 0; t < 16; ++t) {
    const int c0 = 4 * t + 2 * half;
    v2f a; a.x = av[2 * t]; a.y = av[2 * t + 1];
    v2f b0, b1;
    b0.x = lw[n * 65 + c0];         b0.y = lw[n * 65 + c0 + 1];
    b1.x = lw[(n + 16) * 65 + c0];  b1.y = lw[(n + 16) * 65 + c0 + 1];
    acc0 = WMMA_F32X4(a, b0, acc0);
    acc1 = WMMA_F32X4(a, b1, acc1);
  }

  const int mb = half << 3;
  float* outp = out + (size_t)b * 32 * IMG_HW + p0 + mb;
#define K4_STORE(NT, ACC)                                                    \
  {                                                                          \
    const int o = (NT) * 16 + n;                                             \
    const float sh = lsh[o];                                                 \
    _Pragma("unroll") for (int r = 0; r < 8; ++r) {                          \
      float v = (ACC)[r] + sh;                                               \
      outp[(size_t)o * IMG_HW + r] = (v > 0.f) ? v : 0.01f * v;              \
    }                                                                        \
  }
  K4_STORE(0, acc0) K4_STORE(1, acc1)
#undef K4_STORE
}

// ---------------------------------------------------------------------------
// Host launch. Workspace layout (floats): cat[4*64*HW] | pre[4*16*HW] | wgt[4*36*HW]
// = ~121.6 MB total.
// ---------------------------------------------------------------------------
extern "C" void kernel_launch(void* const* d_in, const int* in_sizes, int n_in,
                              void* d_out, int out_size, void* d_ws, size_t ws_size,
                              hipStream_t stream) {
  (void)in_sizes; (void)n_in; (void)out_size; (void)ws_size;
  const float* high      = (const float*)d_in[0];
  const float* low       = (const float*)d_in[1];
  const float* emb       = (const float*)d_in[2];
  const float* reduce_w  = (const float*)d_in[3];
  const float* reduce_b  = (const float*)d_in[4];
  const float* conv1_w   = (const float*)d_in[5];
  const float* conv1_b   = (const float*)d_in[6];
  const float* embconv_w = (const float*)d_in[7];
  const float* embconv_b = (const float*)d_in[8];
  const float* conv2_w   = (const float*)d_in[9];
  const float* conv2_b   = (const float*)d_in[10];
  const float* out_w     = (const float*)d_in[11];
  const float* out_b     = (const float*)d_in[12];
  const float* bn_g      = (const float*)d_in[13];
  const float* bn_b      = (const float*)d_in[14];
  const float* bn_m      = (const float*)d_in[15];
  const float* bn_v      = (const float*)d_in[16];
  float* out = (float*)d_out;

  float* ws_cat = (float*)d_ws;
  float* ws_pre = ws_cat + (size_t)4 * 64 * IMG_HW;
  float* ws_wgt = ws_pre + (size_t)4 * 16 * IMG_HW;

  const dim3 blk(256);
  const dim3 grd(2048);                                // 16384 pixel-tiles / 8 waves
  hipLaunchKernelGGL(k1_reduce,  grd, blk, 0, stream,
                     high, low, reduce_w, reduce_b, ws_cat);
  hipLaunchKernelGGL(k2_pre,     grd, blk, 0, stream,
                     ws_cat, emb, conv1_w, conv1_b, embconv_w, embconv_b, ws_pre);
  hipLaunchKernelGGL(k3_conv3x3, grd, blk, 0, stream,
                     ws_pre, conv2_w, conv2_b, ws_wgt);
  hipLaunchKernelGGL(k4_final,   grd, blk, 0, stream,
                     ws_cat, ws_wgt, out_w, out_b, bn_g, bn_b, bn_m, bn_v, out);
}